// SAOTPrompt_64372969832615
// MI455X (gfx1250) — compile-verified
//
#include <hip/hip_runtime.h>
#include <hip/hip_bf16.h>

#define N_NODES 100000
#define DIM     128
#define NP      64
#define NE      3200000
#define NEPT    5000000
#define INV_EPS 10.0f
#define SINK_IT 20

typedef float v2f __attribute__((ext_vector_type(2)));
typedef float v8f __attribute__((ext_vector_type(8)));

// workspace layout (float offsets)
#define OFF_XA   0ull            // N*D : x_aggr -> xs -> xf (reused)
#define OFF_DEG  12800000ull     // N
#define OFF_DIS  12900000ull     // N
#define OFF_U    13000000ull     // N
#define OFF_KTU  13100000ull     // 64
#define OFF_VV   13100064ull     // 64
#define OFF_WMAX 13100128ull     // 1 (uint bits)
#define OFF_PN   13100160ull     // P*D (16B aligned)
#define OFF_S    13108352ull     // N*P : S -> T (reused)

__device__ __forceinline__ float warp_sum(float v) {
    #pragma unroll
    for (int m = 16; m > 0; m >>= 1) v += __shfl_xor(v, m, 32);
    return v;
}

// ---------------- init: zero accumulators, set u = 1/N ----------------
__global__ void init_kernel(float* __restrict__ xa, float* __restrict__ deg,
                            float* __restrict__ u, float* __restrict__ ktu,
                            unsigned* __restrict__ wmax, float* __restrict__ loss) {
    size_t idx = (size_t)blockIdx.x * 256 + threadIdx.x;
    size_t total = (size_t)N_NODES * DIM;
    for (size_t i = idx; i < total; i += (size_t)gridDim.x * 256) xa[i] = 0.f;
    if (idx < N_NODES) { deg[idx] = 0.f; u[idx] = 1.0f / (float)N_NODES; }
    if (idx < 64) ktu[idx] = 0.f;
    if (idx == 0) { *wmax = 0u; *loss = 0.f; }
}

// ---------------- degree (count of col) ----------------
__global__ void deg_kernel(const int* __restrict__ col, float* __restrict__ deg) {
    int e = blockIdx.x * 256 + threadIdx.x;
    if (e < NE) atomicAdd(&deg[col[e]], 1.0f);
}

__global__ void dis_kernel(const float* __restrict__ deg, float* __restrict__ dis) {
    int i = blockIdx.x * 256 + threadIdx.x;
    if (i < N_NODES) { float d = deg[i]; dis[i] = (d > 0.f) ? (1.0f / sqrtf(d)) : 0.f; }
}

// ---------------- scatter aggregation: xa[row] += x[col]*nw, wave/edge ----------------
__global__ void aggr_kernel(const float* __restrict__ x, const int* __restrict__ row,
                            const int* __restrict__ col, const float* __restrict__ ew,
                            const float* __restrict__ dis, float* __restrict__ xa) {
    int e    = (int)(((size_t)blockIdx.x * 256 + threadIdx.x) >> 5);
    int lane = threadIdx.x & 31;
    if (e >= NE) return;
    int r = row[e], c = col[e];
    float nw = dis[r] * ew[e] * dis[c];
    float4 xc = ((const float4*)(x + (size_t)c * DIM))[lane];
    float* dst = xa + (size_t)r * DIM + lane * 4;
    atomicAdd(dst + 0, xc.x * nw);
    atomicAdd(dst + 1, xc.y * nw);
    atomicAdd(dst + 2, xc.z * nw);
    atomicAdd(dst + 3, xc.w * nw);
}

// ---------------- x_struct = (1-g)x + g*xa, l2norm row, in-place over xa ----------------
__global__ void structnorm_kernel(const float* __restrict__ x, float* __restrict__ xa,
                                  const float* __restrict__ gamma_p) {
    int wid = threadIdx.x >> 5, lane = threadIdx.x & 31;
    int rowi = blockIdx.x * 8 + wid;
    if (rowi >= N_NODES) return;
    float g = *gamma_p;
    size_t base = (size_t)rowi * DIM;
    float4 xv = ((const float4*)(x + base))[lane];
    float4 av = ((const float4*)(xa + base))[lane];
    float4 s;
    s.x = (1.f - g) * xv.x + g * av.x;
    s.y = (1.f - g) * xv.y + g * av.y;
    s.z = (1.f - g) * xv.z + g * av.z;
    s.w = (1.f - g) * xv.w + g * av.w;
    float ss = warp_sum(s.x * s.x + s.y * s.y + s.z * s.z + s.w * s.w);
    float inv = 1.f / fmaxf(sqrtf(ss), 1e-12f);
    s.x *= inv; s.y *= inv; s.z *= inv; s.w *= inv;
    ((float4*)(xa + base))[lane] = s;
}

// ---------------- normalize prompt tokens -> pn ----------------
__global__ void pnorm_kernel(const float* __restrict__ pt, float* __restrict__ pn) {
    int wid = threadIdx.x >> 5, lane = threadIdx.x & 31;
    int rowi = blockIdx.x * 8 + wid;
    if (rowi >= NP) return;
    size_t base = (size_t)rowi * DIM;
    float4 s = ((const float4*)(pt + base))[lane];
    float ss = warp_sum(s.x * s.x + s.y * s.y + s.z * s.z + s.w * s.w);
    float inv = 1.f / fmaxf(sqrtf(ss), 1e-12f);
    s.x *= inv; s.y *= inv; s.z *= inv; s.w *= inv;
    ((float4*)(pn + base))[lane] = s;
}

// ---------------- S = xs @ pn^T via V_WMMA_F32_16X16X4_F32 ----------------
__global__ __launch_bounds__(256) void s_wmma_kernel(const float* __restrict__ xs,
                                                     const float* __restrict__ pn,
                                                     float* __restrict__ S) {
    __shared__ float lpn[NP * DIM];
    for (int i = threadIdx.x; i < NP * DIM / 4; i += 256)
        ((float4*)lpn)[i] = ((const float4*)pn)[i];
    __syncthreads();
    int wid = threadIdx.x >> 5, lane = threadIdx.x & 31;
    int tile = blockIdx.x * 8 + wid;
    if (tile >= N_NODES / 16) return;
    int n = lane & 15, h = lane >> 4;           // lane = 16*h + n
    const float* arow = xs + (size_t)(tile * 16 + n) * DIM + 2 * h;
    v8f acc[4] = {};
    for (int kb = 0; kb < DIM; kb += 4) {       // A[m][k=v+2h] ; B[k][n] = pn[n][k]
        v2f a; a.x = arow[kb]; a.y = arow[kb + 1];
        #pragma unroll
        for (int nt = 0; nt < 4; nt++) {
            const float* bp = lpn + (size_t)(nt * 16 + n) * DIM + kb + 2 * h;
            v2f b; b.x = bp[0]; b.y = bp[1];
            acc[nt] = __builtin_amdgcn_wmma_f32_16x16x4_f32(
                false, a, false, b, (short)0, acc[nt], false, false);
        }
    }
    #pragma unroll
    for (int nt = 0; nt < 4; nt++)
        #pragma unroll
        for (int r = 0; r < 8; r++)             // D[r+8h][n]
            S[(size_t)(tile * 16 + r + 8 * h) * NP + nt * 16 + n] = acc[nt][r];
}

// ---------------- Sinkhorn column pass: ktu[j] += sum_i K(i,j)*u[i] ----------------
__global__ void col_kernel(const float* __restrict__ S, const float* __restrict__ u,
                           float* __restrict__ ktu) {
    __shared__ float lds[64];
    if (threadIdx.x < 64) lds[threadIdx.x] = 0.f;
    __syncthreads();
    int lane = threadIdx.x & 31;
    int warpG = (int)(((size_t)blockIdx.x * 256 + threadIdx.x) >> 5);
    int nwarp = (int)(((size_t)gridDim.x * 256) >> 5);
    float s0 = 0.f, s1 = 0.f;
    for (int i = warpG; i < N_NODES; i += nwarp) {
        float ui = u[i];
        const float* row = S + (size_t)i * NP;
        s0 += expf((row[lane]      - 1.f) * INV_EPS) * ui;
        s1 += expf((row[lane + 32] - 1.f) * INV_EPS) * ui;
    }
    atomicAdd(&lds[lane], s0);
    atomicAdd(&lds[lane + 32], s1);
    __syncthreads();
    if (threadIdx.x < 64) atomicAdd(&ktu[threadIdx.x], lds[threadIdx.x]);
}

__global__ void vfix_kernel(float* __restrict__ ktu, float* __restrict__ vv) {
    int t = threadIdx.x;
    if (t < 64) { vv[t] = (1.0f / (float)NP) / (ktu[t] + 1e-30f); ktu[t] = 0.f; }
}

// ---------------- Sinkhorn row pass: u[i] = a / (K(i,:).v + eps) ----------------
__global__ void row_kernel(const float* __restrict__ S, const float* __restrict__ vv,
                           float* __restrict__ u) {
    int lane = threadIdx.x & 31;
    int i = (int)(((size_t)blockIdx.x * 256 + threadIdx.x) >> 5);
    if (i >= N_NODES) return;
    float v0 = vv[lane], v1 = vv[lane + 32];
    const float* row = S + (size_t)i * NP;
    float s = expf((row[lane] - 1.f) * INV_EPS) * v0
            + expf((row[lane + 32] - 1.f) * INV_EPS) * v1;
    s = warp_sum(s);
    if (lane == 0) u[i] = (1.0f / (float)N_NODES) / (s + 1e-30f);
}

// ---------------- T = u*K*v in place over S ; ot_loss += T*(1-S) ----------------
__global__ void tloss_kernel(float* __restrict__ S, const float* __restrict__ u,
                             const float* __restrict__ vv, float* __restrict__ loss) {
    float part = 0.f;
    size_t total = (size_t)N_NODES * NP;
    for (size_t idx = (size_t)blockIdx.x * 256 + threadIdx.x; idx < total;
         idx += (size_t)gridDim.x * 256) {
        int i = (int)(idx >> 6), j = (int)(idx & 63);
        float s = S[idx];
        float k = expf((s - 1.f) * INV_EPS);
        float t = u[i] * k * vv[j];
        part += t * (1.f - s);
        S[idx] = t;
    }
    part = warp_sum(part);
    __shared__ float lw[8];
    int wid = threadIdx.x >> 5, lane = threadIdx.x & 31;
    if (lane == 0) lw[wid] = part;
    __syncthreads();
    if (threadIdx.x == 0) {
        float s = 0.f;
        #pragma unroll
        for (int i = 0; i < 8; i++) s += lw[i];
        atomicAdd(loss, s);
    }
}

// ---------------- x_adapted = x + alpha*N*(T @ prompt_tokens) via WMMA ----------------
__global__ __launch_bounds__(256) void pm_wmma_kernel(const float* __restrict__ T,
                                                      const float* __restrict__ pt,
                                                      const float* __restrict__ x,
                                                      const float* __restrict__ alpha_p,
                                                      float* __restrict__ out) {
    __shared__ float lpt[NP * DIM];
    for (int i = threadIdx.x; i < NP * DIM / 4; i += 256)
        ((float4*)lpt)[i] = ((const float4*)pt)[i];
    __syncthreads();
    int wid = threadIdx.x >> 5, lane = threadIdx.x & 31;
    int tile = blockIdx.x * 8 + wid;
    if (tile >= N_NODES / 16) return;
    int n = lane & 15, h = lane >> 4;
    float aN = (*alpha_p) * (float)N_NODES;
    const float* arow = T + (size_t)(tile * 16 + n) * NP + 2 * h;
    v8f acc[8] = {};
    for (int kb = 0; kb < NP; kb += 4) {        // A = T[m][k], B[k][n] = pt[k][col]
        v2f a; a.x = arow[kb]; a.y = arow[kb + 1];
        #pragma unroll
        for (int nt = 0; nt < 8; nt++) {
            const float* bp = lpt + (size_t)(kb + 2 * h) * DIM + nt * 16 + n;
            v2f b; b.x = bp[0]; b.y = bp[DIM];
            acc[nt] = __builtin_amdgcn_wmma_f32_16x16x4_f32(
                false, a, false, b, (short)0, acc[nt], false, false);
        }
    }
    #pragma unroll
    for (int nt = 0; nt < 8; nt++)
        #pragma unroll
        for (int r = 0; r < 8; r++) {
            size_t o = (size_t)(tile * 16 + r + 8 * h) * DIM + nt * 16 + n;
            out[o] = x[o] + aN * acc[nt][r];
        }
}

// ---------------- xf = l2norm(x_adapted) ----------------
__global__ void xf_kernel(const float* __restrict__ xad, float* __restrict__ xf) {
    int wid = threadIdx.x >> 5, lane = threadIdx.x & 31;
    int rowi = blockIdx.x * 8 + wid;
    if (rowi >= N_NODES) return;
    size_t base = (size_t)rowi * DIM;
    float4 s = ((const float4*)(xad + base))[lane];
    float ss = warp_sum(s.x * s.x + s.y * s.y + s.z * s.z + s.w * s.w);
    float inv = 1.f / fmaxf(sqrtf(ss), 1e-12f);
    s.x *= inv; s.y *= inv; s.z *= inv; s.w *= inv;
    ((float4*)(xf + base))[lane] = s;
}

// ---------------- kNN edge scoring: w = relu(dot), track global max ----------------
__global__ void edge_kernel(const float* __restrict__ xf, const int* __restrict__ er,
                            const int* __restrict__ ec, float* __restrict__ wout,
                            unsigned* __restrict__ wmax) {
    int wid = threadIdx.x >> 5, lane = threadIdx.x & 31;
    size_t e = (size_t)blockIdx.x * 8 + wid;
    float w = 0.f;
    if (e < NEPT) {
        int a = er[e], b = ec[e];
        float4 xa4 = ((const float4*)(xf + (size_t)a * DIM))[lane];
        float4 xb4 = ((const float4*)(xf + (size_t)b * DIM))[lane];
        float d = xa4.x * xb4.x + xa4.y * xb4.y + xa4.z * xb4.z + xa4.w * xb4.w;
        d = warp_sum(d);
        w = fmaxf(d, 0.f);
        if (lane == 0) wout[e] = w;
    }
    __shared__ float lm[8];
    if (lane == 0) lm[wid] = w;
    __syncthreads();
    if (threadIdx.x == 0) {
        float m = lm[0];
        #pragma unroll
        for (int i = 1; i < 8; i++) m = fmaxf(m, lm[i]);
        atomicMax(wmax, __float_as_uint(m));   // valid: all values >= 0
    }
}

__global__ void wnorm_kernel(float* __restrict__ wout, const unsigned* __restrict__ wmax) {
    float scale = 1.0f / (__uint_as_float(*wmax) + 1e-8f);
    for (size_t e = (size_t)blockIdx.x * 256 + threadIdx.x; e < NEPT;
         e += (size_t)gridDim.x * 256)
        wout[e] *= scale;
}

extern "C" void kernel_launch(void* const* d_in, const int* in_sizes, int n_in,
                              void* d_out, int out_size, void* d_ws, size_t ws_size,
                              hipStream_t stream) {
    const float* x       = (const float*)d_in[0];
    const int*   ei      = (const int*)  d_in[1];   // [2,E]: row then col
    const float* ew      = (const float*)d_in[2];
    const int*   pei     = (const int*)  d_in[3];   // [2,EPT]
    const float* pt      = (const float*)d_in[4];
    const float* alpha_p = (const float*)d_in[5];
    const float* gamma_p = (const float*)d_in[6];

    float* out  = (float*)d_out;
    float* ws   = (float*)d_ws;
    float* xa   = ws + OFF_XA;     // also xs, then xf
    float* deg  = ws + OFF_DEG;
    float* dis  = ws + OFF_DIS;
    float* u    = ws + OFF_U;
    float* ktu  = ws + OFF_KTU;
    float* vv   = ws + OFF_VV;
    unsigned* wmax = (unsigned*)(ws + OFF_WMAX);
    float* pn   = ws + OFF_PN;
    float* S    = ws + OFF_S;      // also T

    float* loss = out + (size_t)N_NODES * DIM;
    float* wout = loss + 1;

    init_kernel<<<50000, 256, 0, stream>>>(xa, deg, u, ktu, wmax, loss);
    deg_kernel<<<(NE + 255) / 256, 256, 0, stream>>>(ei + NE, deg);
    dis_kernel<<<(N_NODES + 255) / 256, 256, 0, stream>>>(deg, dis);
    aggr_kernel<<<NE / 8, 256, 0, stream>>>(x, ei, ei + NE, ew, dis, xa);
    structnorm_kernel<<<N_NODES / 8, 256, 0, stream>>>(x, xa, gamma_p);
    pnorm_kernel<<<8, 256, 0, stream>>>(pt, pn);
    s_wmma_kernel<<<(N_NODES / 16 + 7) / 8, 256, 0, stream>>>(xa, pn, S);

    for (int it = 0; it < SINK_IT; it++) {
        col_kernel<<<1024, 256, 0, stream>>>(S, u, ktu);
        vfix_kernel<<<1, 64, 0, stream>>>(ktu, vv);
        row_kernel<<<N_NODES / 8, 256, 0, stream>>>(S, vv, u);
    }

    tloss_kernel<<<4096, 256, 0, stream>>>(S, u, vv, loss);
    pm_wmma_kernel<<<(N_NODES / 16 + 7) / 8, 256, 0, stream>>>(S, pt, x, alpha_p, out);
    xf_kernel<<<N_NODES / 8, 256, 0, stream>>>(out, xa);
    edge_kernel<<<NEPT / 8, 256, 0, stream>>>(xa, pei, pei + NEPT, wout, wmax);
    wnorm_kernel<<<8192, 256, 0, stream>>>(wout, wmax);
}